// AutoFM_8022998909053
// MI455X (gfx1250) — compile-verified
//
#include <hip/hip_runtime.h>

// AutoFM forward for MI455X (gfx1250, wave32, WMMA).
// B=1024, F=39, K=16, P=741 pairs, T=9139 triples.

#define B_   1024
#define F_   39
#define K_   16
#define P_   741
#define T_   9139
#define FP_  48          // F padded to 3 tiles of 16
#define PP_  752         // P padded to 47 tiles of 16
#define EPS_ 1e-3f

typedef float v2f __attribute__((ext_vector_type(2)));
typedef float v8f __attribute__((ext_vector_type(8)));

// ---------------------------------------------------------------------------
// K1: gather first-order weights and embeddings; pad embedding rows to 48.
// Grid: B blocks x 64 threads.
__global__ void k_gather(const int* __restrict__ feats, const float* __restrict__ w,
                         const float* __restrict__ v, float* __restrict__ Xg,
                         float* __restrict__ lin) {
    __shared__ int   fl[F_];
    __shared__ float red[64];
    const int b = blockIdx.x, tid = threadIdx.x;
    if (tid < F_) fl[tid] = feats[b * F_ + tid];
    __syncthreads();
    red[tid] = (tid < F_) ? w[fl[tid]] : 0.f;
    __syncthreads();
    for (int s = 32; s > 0; s >>= 1) {
        if (tid < s) red[tid] += red[tid + s];
        __syncthreads();
    }
    if (tid == 0) lin[b] = red[0];
    for (int idx = tid; idx < FP_ * K_; idx += 64) {
        const int r = idx >> 4, k = idx & 15;
        const float val = (r < F_) ? v[(long)fl[r] * K_ + k] : 0.f;
        Xg[(long)b * (FP_ * K_) + idx] = val;
    }
}

// ---------------------------------------------------------------------------
// K2: per-sample interactions.
//   H[p,k]   = X[i1(p),k] * X[i2(p),k]          (741x16, zero-padded to 752)
//   level2[p]= sum_k H[p,k]                      (row sums)
//   level3   = H (752x16) * X^T (16x48) via v_wmma_f32_16x16x4_f32, valid
//              (p, j) entries scattered to L3[b, t] with closed-form t(p,j).
// Grid: B blocks x 256 threads (8 waves).
__global__ void __launch_bounds__(256) k_interact(
        const float* __restrict__ Xg, const int* __restrict__ rows,
        const int* __restrict__ cols, float* __restrict__ L2,
        float* __restrict__ L3) {
    __shared__ __align__(16) float Xl[FP_ * K_];   //  3 KB
    __shared__ __align__(16) float Hl[PP_ * K_];   // 48 KB
    __shared__ int rl[P_], cl[P_], tl[P_];         //  ~9 KB
    const int b = blockIdx.x, tid = threadIdx.x;

    for (int i = tid; i < FP_ * K_; i += 256)
        Xl[i] = Xg[(long)b * (FP_ * K_) + i];

    for (int p = tid; p < P_; p += 256) {
        const int i = cols[p], jj = rows[p];       // pair (i, jj), i < jj
        rl[p] = jj; cl[p] = i;
        // #triples lexicographically before pair (i, jj):
        const int n39  = 39 - i;
        const int base = T_ - (n39 * (n39 - 1) * (n39 - 2)) / 6;   // i1 < i
        const int sec  = ((37 - i) * (38 - i)) / 2
                       - ((38 - jj) * (39 - jj)) / 2;              // i1==i, i2<jj
        tl[p] = base + sec;                        // t(p,j) = tl[p] + j - jj - 1
    }
    __syncthreads();

    for (int idx = tid; idx < PP_ * K_; idx += 256) {
        const int p = idx >> 4, k = idx & 15;
        Hl[idx] = (p < P_) ? Xl[rl[p] * K_ + k] * Xl[cl[p] * K_ + k] : 0.f;
    }
    __syncthreads();

    // level2 = row sums of H
    for (int p = tid; p < P_; p += 256) {
        float s = 0.f;
        #pragma unroll
        for (int k = 0; k < K_; ++k) s += Hl[p * K_ + k];
        L2[(long)b * P_ + p] = s;
    }

    // level3 GEMM: 47x3 = 141 output tiles of 16x16, K=16 as 4 chained K=4 WMMAs.
    const int wave = tid >> 5, lane = tid & 31;
    const int half = lane >> 4, lm = lane & 15;
    const int NT = (PP_ / 16) * (FP_ / 16);        // 141
    for (int tile = wave; tile < NT; tile += 8) {
        const int mt = tile / 3, nt = tile % 3;
        v8f acc = {};
        #pragma unroll
        for (int c = 0; c < 4; ++c) {
            const int k0 = 4 * c + 2 * half;       // even -> 8B aligned float2
            v2f a  = *(const v2f*)&Hl[(mt * 16 + lm) * K_ + k0];  // A: lane=M
            v2f bb = *(const v2f*)&Xl[(nt * 16 + lm) * K_ + k0];  // B: lane=N
            acc = __builtin_amdgcn_wmma_f32_16x16x4_f32(
                false, a, false, bb, (short)0, acc, false, false);
        }
        // D layout: VGPR g, lanes 0-15 -> M=g, lanes 16-31 -> M=g+8; N=lane%16
        #pragma unroll
        for (int g = 0; g < 8; ++g) {
            const int p = mt * 16 + g + 8 * half;
            const int j = nt * 16 + lm;
            if (p < P_ && j < F_) {
                const int jj = rl[p];
                if (j > jj) {
                    const int t = tl[p] + (j - jj - 1);
                    L3[(long)b * T_ + t] = acc[g];
                }
            }
        }
    }
}

// ---------------------------------------------------------------------------
// K3: per-column batch stats -> fold BN * gw into affine (a, c) per column.
__global__ void k_stats(const float* __restrict__ M, int C,
                        const float* __restrict__ gamma,
                        const float* __restrict__ beta,
                        const float* __restrict__ gw,
                        float* __restrict__ a, float* __restrict__ cc) {
    const int p = blockIdx.x * blockDim.x + threadIdx.x;
    if (p >= C) return;
    float s = 0.f, q = 0.f;
    for (int b = 0; b < B_; ++b) {
        const float x = M[(long)b * C + p];
        s += x; q += x * x;
    }
    const float m   = s * (1.f / B_);
    const float var = fmaxf(q * (1.f / B_) - m * m, 0.f);
    const float rs  = rsqrtf(var + EPS_);
    const float g   = gamma[p] * rs;
    a[p]  = g * gw[p];
    cc[p] = gw[p] * (beta[p] - m * g);
}

// ---------------------------------------------------------------------------
// K4: logits[b] = lin[b] + bias + sum_p (a2*L2 + c2) + sum_t (a3*L3 + c3)
__global__ void __launch_bounds__(256) k_final(
        const float* __restrict__ L2, const float* __restrict__ L3,
        const float* __restrict__ a2, const float* __restrict__ c2,
        const float* __restrict__ a3, const float* __restrict__ c3,
        const float* __restrict__ lin, const float* __restrict__ bias,
        float* __restrict__ out) {
    __shared__ float red[256];
    const int b = blockIdx.x, tid = threadIdx.x;
    float acc = 0.f;
    for (int p = tid; p < P_; p += 256) acc += fmaf(a2[p], L2[(long)b * P_ + p], c2[p]);
    for (int t = tid; t < T_; t += 256) acc += fmaf(a3[t], L3[(long)b * T_ + t], c3[t]);
    red[tid] = acc;
    __syncthreads();
    for (int s = 128; s > 0; s >>= 1) {
        if (tid < s) red[tid] += red[tid + s];
        __syncthreads();
    }
    if (tid == 0) out[b] = red[0] + lin[b] + bias[0];
}

// ---------------------------------------------------------------------------
extern "C" void kernel_launch(void* const* d_in, const int* in_sizes, int n_in,
                              void* d_out, int out_size, void* d_ws, size_t ws_size,
                              hipStream_t stream) {
    (void)in_sizes; (void)n_in; (void)out_size; (void)ws_size;
    const int*   feats = (const int*)  d_in[0];
    const float* w     = (const float*)d_in[1];
    const float* v     = (const float*)d_in[2];
    const float* bias  = (const float*)d_in[3];
    const float* g2    = (const float*)d_in[4];
    const float* be2   = (const float*)d_in[5];
    const float* gw2   = (const float*)d_in[6];
    const float* g3    = (const float*)d_in[7];
    const float* be3   = (const float*)d_in[8];
    const float* gw3   = (const float*)d_in[9];
    const int*   rows  = (const int*)  d_in[10];
    const int*   cols  = (const int*)  d_in[11];
    // d_in[12..14] = i1,i2,i3 (unused: t(p,j) computed in closed form)

    float* ws  = (float*)d_ws;          // ~43.7 MB used
    size_t off = 0;
    float* Xg  = ws + off; off += (size_t)B_ * FP_ * K_;
    float* L2  = ws + off; off += (size_t)B_ * P_;
    float* L3  = ws + off; off += (size_t)B_ * T_;
    float* lin = ws + off; off += B_;
    float* a2  = ws + off; off += P_;
    float* c2  = ws + off; off += P_;
    float* a3  = ws + off; off += T_;
    float* c3  = ws + off; off += T_;

    k_gather  <<<B_, 64, 0, stream>>>(feats, w, v, Xg, lin);
    k_interact<<<B_, 256, 0, stream>>>(Xg, rows, cols, L2, L3);
    k_stats   <<<(P_ + 255) / 256, 256, 0, stream>>>(L2, P_, g2, be2, gw2, a2, c2);
    k_stats   <<<(T_ + 255) / 256, 256, 0, stream>>>(L3, T_, g3, be3, gw3, a3, c3);
    k_final   <<<B_, 256, 0, stream>>>(L2, L3, a2, c2, a3, c3, lin, bias, (float*)d_out);
}